// TopKSimilarityLayer_21741124452844
// MI455X (gfx1250) — compile-verified
//
#include <hip/hip_runtime.h>

typedef __attribute__((ext_vector_type(2))) float v2f;
typedef __attribute__((ext_vector_type(8))) float v8f;

#define NQ   4096   // queries
#define NR   4096   // table rows
#define NV   128    // values per row
#define ND   128    // vector dim
#define KTOP 16

// ---------------------------------------------------------------------------
// Phase 1: scores GEMM + per-row max/argmax.
// Block = 256 threads = 8 waves. Wave w handles table row (blockIdx.x*8 + w)
// for the 16 queries starting at blockIdx.y*16.
// Per wave: C tiles are 16(queries) x 16(values); 8 N-tiles cover the 128
// values; K=128 in 32 steps of v_wmma_f32_16x16x4_f32 (256 WMMA per wave).
// B fragments are double-buffered in 4-step groups and the accumulation is
// split into two chains to keep the XDL pipe fed.
// ---------------------------------------------------------------------------
__global__ __launch_bounds__(256) void topk_phase1(
    const float* __restrict__ q, const float* __restrict__ tbl,
    float* __restrict__ rowmax, unsigned char* __restrict__ rowarg)
{
  const int lane  = threadIdx.x & 31;
  const int wave  = threadIdx.x >> 5;
  const int r     = blockIdx.x * 8 + wave;
  const int qbase = blockIdx.y * 16;
  const int m     = lane & 15;           // M row (both half-waves)
  const int khalf = (lane >> 4) << 1;    // 0 for lo half, 2 for hi half

  // A fragments for all 32 K-steps (ISA 16x4 f32 A layout):
  //   lane<16 : (K=4s+0, K=4s+1) for row M=lane
  //   lane>=16: (K=4s+2, K=4s+3) for row M=lane-16
  v2f a[32];
  const float* qp = q + (size_t)(qbase + m) * ND + khalf;
#pragma unroll
  for (int s = 0; s < 32; ++s) {
    a[s].x = qp[4 * s + 0];
    a[s].y = qp[4 * s + 1];
  }

  float best[8];
  int   bidx[8];
#pragma unroll
  for (int i = 0; i < 8; ++i) { best[i] = -3.402823466e38f; bidx[i] = 0; }

  const float* trow = tbl + (size_t)r * NV * ND;

  for (int t = 0; t < 8; ++t) {
    const int n = t * 16 + m;                      // value index (N column)
    const float* bp = trow + (size_t)n * ND + khalf;

    v8f acc0 = {0.f, 0.f, 0.f, 0.f, 0.f, 0.f, 0.f, 0.f};
    v8f acc1 = {0.f, 0.f, 0.f, 0.f, 0.f, 0.f, 0.f, 0.f};

    // Double-buffered B fragments, groups of 4 K-steps.
    v2f bb[2][4];
#pragma unroll
    for (int j = 0; j < 4; ++j) {
      bb[0][j].x = bp[4 * j + 0];
      bb[0][j].y = bp[4 * j + 1];
    }
#pragma unroll
    for (int g = 0; g < 8; ++g) {
      const int cur = g & 1, nxt = cur ^ 1;
      if (g < 7) {                                 // prefetch next group
#pragma unroll
        for (int j = 0; j < 4; ++j) {
          const int s = 4 * (g + 1) + j;
          bb[nxt][j].x = bp[4 * s + 0];
          bb[nxt][j].y = bp[4 * s + 1];
        }
      }
#pragma unroll
      for (int j = 0; j < 4; ++j) {
        const int s = 4 * g + j;
        if (j & 1)
          acc1 = __builtin_amdgcn_wmma_f32_16x16x4_f32(
              false, a[s], false, bb[cur][j], (short)0, acc1, false, false);
        else
          acc0 = __builtin_amdgcn_wmma_f32_16x16x4_f32(
              false, a[s], false, bb[cur][j], (short)0, acc0, false, false);
      }
    }

    // C layout: vgpr i -> query (i + 8*(lane>>4)), column n = t*16 + (lane&15)
#pragma unroll
    for (int i = 0; i < 8; ++i) {
      float v = acc0[i] + acc1[i];
      if (v > best[i]) { best[i] = v; bidx[i] = n; }   // first-max on ties
    }
  }

  // Max/argmax across the 16 lanes of each half-wave (masks stay in-half).
#pragma unroll
  for (int mask = 1; mask <= 8; mask <<= 1) {
#pragma unroll
    for (int i = 0; i < 8; ++i) {
      float ov = __shfl_xor(best[i], mask, 32);
      int   oi = __shfl_xor(bidx[i], mask, 32);
      if (ov > best[i] || (ov == best[i] && oi < bidx[i])) {
        best[i] = ov; bidx[i] = oi;
      }
    }
  }

  if (m == 0) {                       // lane 0 and lane 16 write
    const int mofs = (lane >> 4) * 8;
#pragma unroll
    for (int i = 0; i < 8; ++i) {
      const int qi = qbase + mofs + i;
      rowmax[(size_t)qi * NR + r] = best[i];
      rowarg[(size_t)qi * NR + r] = (unsigned char)bidx[i];
    }
  }
}

// ---------------------------------------------------------------------------
// Phase 2: per query, top-16 rows by max score; gather vectors/scores/ids.
// One block (256 threads) per query; 16 block-wide argmax passes w/ exclusion.
// ---------------------------------------------------------------------------
__global__ __launch_bounds__(256) void topk_phase2(
    const float* __restrict__ tbl, const float* __restrict__ rowmax,
    const unsigned char* __restrict__ rowarg, float* __restrict__ out)
{
  __shared__ float sval[256];
  __shared__ int   srow[256];
  __shared__ int   topr[KTOP];
  __shared__ float topv[KTOP];

  const int q   = blockIdx.x;
  const int tid = threadIdx.x;

  float s[16];
#pragma unroll
  for (int j = 0; j < 16; ++j)
    s[j] = rowmax[(size_t)q * NR + tid + 256 * j];

  for (int pick = 0; pick < KTOP; ++pick) {
    float bv = -3.402823466e38f; int br = 0x7fffffff;
#pragma unroll
    for (int j = 0; j < 16; ++j) {
      int r = tid + 256 * j;
      if (s[j] > bv || (s[j] == bv && r < br)) { bv = s[j]; br = r; }
    }
    sval[tid] = bv; srow[tid] = br;
    __syncthreads();
    for (int off = 128; off > 0; off >>= 1) {
      if (tid < off) {
        float ov = sval[tid + off]; int orr = srow[tid + off];
        if (ov > sval[tid] || (ov == sval[tid] && orr < srow[tid])) {
          sval[tid] = ov; srow[tid] = orr;
        }
      }
      __syncthreads();
    }
    int wr = srow[0]; float wv = sval[0];
    if (tid == 0) { topr[pick] = wr; topv[pick] = wv; }
    if ((wr & 255) == tid) s[wr >> 8] = -3.402823466e38f;  // exclude winner
    __syncthreads();
  }

  float* out_vals   = out;                               // [NQ,KTOP,ND]
  float* out_scores = out + (size_t)NQ * KTOP * ND;      // [NQ,KTOP]
  float* out_ids    = out_scores + (size_t)NQ * KTOP;    // [NQ,KTOP]

  if (tid < KTOP) {
    int r = topr[tid];
    int a = rowarg[(size_t)q * NR + r];
    out_scores[(size_t)q * KTOP + tid] = topv[tid];
    out_ids[(size_t)q * KTOP + tid]    = (float)(r * NV + a);
  }
  for (int idx = tid; idx < KTOP * ND; idx += 256) {
    int k = idx >> 7, d = idx & 127;
    int r = topr[k];
    int a = rowarg[(size_t)q * NR + r];
    out_vals[((size_t)q * KTOP + k) * ND + d] =
        tbl[((size_t)r * NV + a) * ND + d];
  }
}

// ---------------------------------------------------------------------------
extern "C" void kernel_launch(void* const* d_in, const int* in_sizes, int n_in,
                              void* d_out, int out_size, void* d_ws, size_t ws_size,
                              hipStream_t stream) {
  (void)in_sizes; (void)n_in; (void)out_size; (void)ws_size;
  const float* queries = (const float*)d_in[0];
  const float* table   = (const float*)d_in[1];

  float*         rowmax = (float*)d_ws;                               // 64 MB
  unsigned char* rowarg = (unsigned char*)d_ws + (size_t)NQ * NR * 4; // 16 MB

  topk_phase1<<<dim3(NR / 8, NQ / 16), 256, 0, stream>>>(queries, table,
                                                         rowmax, rowarg);
  topk_phase2<<<NQ, 256, 0, stream>>>(table, rowmax, rowarg, (float*)d_out);
}